// comp_encoding_7705171329545
// MI455X (gfx1250) — compile-verified
//
#include <hip/hip_runtime.h>

// ---- static problem dims (match reference) ----
#define BB   512   // molecules
#define AA   128   // atoms / molecule
#define EE   256   // directed bonds / molecule
#define KK   6     // incoming bonds gathered
#define AF   133   // ATOM_FDIM
#define FD   147   // ATOM_FDIM + BOND_FDIM
#define HH   128   // hidden
#define K1P  148   // FD padded to mult of 4 (zero-filled)
#define S2   132   // LDS row stride for H=128 (bank-conflict pad, never read)
#define CATD 261   // AF + HH
#define CATP 268   // CATD padded to mult of 4 + bank pad (zero-filled)

typedef float v2f __attribute__((ext_vector_type(2)));
typedef float v8f __attribute__((ext_vector_type(8)));

// D = A(16x4 f32) * B(4x16 f32) + C(16x16 f32), full fp32 WMMA
__device__ __forceinline__ v8f wmma4(v2f a, v2f b, v8f c) {
  return __builtin_amdgcn_wmma_f32_16x16x4_f32(false, a, false, b, (short)0, c,
                                               false, false);
}

// Copy a [rows x cols] row-major matrix (16B-aligned base, rows*cols % 4 == 0)
// into LDS with row stride `stride`, using b128 global loads with row-straddle
// fixup.  Does NOT zero pad columns.
__device__ __forceinline__ void stage_rows4(const float* __restrict__ src,
                                            float* __restrict__ dst,
                                            int rows, int cols, int stride,
                                            int tid) {
  const int total4 = (rows * cols) >> 2;
  const float4* s4 = (const float4*)src;
  for (int i = tid; i < total4; i += 256) {
    float4 v = s4[i];
    int j = i << 2;
    int r = j / cols, c = j - r * cols;
    if (c + 4 <= cols) {
      float* d = dst + r * stride + c;
      d[0] = v.x; d[1] = v.y; d[2] = v.z; d[3] = v.w;
    } else {  // vector straddles a row boundary (cols not mult of 4)
      float tmp0 = v.x, tmp1 = v.y, tmp2 = v.z, tmp3 = v.w;
      float t[4] = {tmp0, tmp1, tmp2, tmp3};
#pragma unroll
      for (int u = 0; u < 4; ++u) {
        int jj = j + u;
        int rr = jj / cols, cc = jj - rr * cols;
        dst[rr * stride + cc] = t[u];
      }
    }
  }
}

// Zero LDS pad columns [cols, cols+padc) of a [rows x stride] tile.
__device__ __forceinline__ void zero_pad(float* __restrict__ dst, int rows,
                                         int cols, int stride, int padc,
                                         int tid) {
  for (int i = tid; i < rows * padc; i += 256) {
    int r = i / padc, c = i - r * padc;
    dst[r * stride + cols + c] = 0.f;
  }
}

// ---------------------------------------------------------------------------
// Kernel 0: serial prefix sum of mol_lens (512 ints) -> prefix[0..512]
// ---------------------------------------------------------------------------
__global__ void k_prefix(const int* __restrict__ lens, int* __restrict__ prefix) {
  if (threadIdx.x == 0 && blockIdx.x == 0) {
    int acc = 0;
    for (int i = 0; i < BB; ++i) { prefix[i] = acc; acc += lens[i]; }
    prefix[BB] = acc;
  }
}

// ---------------------------------------------------------------------------
// Kernel 1: inp = f_ini @ W_i^T ; msg = relu(inp).  One block per molecule.
// LDS: f tile 256x148 + W_i 128x148 (zero-padded K col 147)
// ---------------------------------------------------------------------------
__global__ void __launch_bounds__(256) k_init(const float* __restrict__ fini,
                                              const float* __restrict__ Wi,
                                              float* __restrict__ inp,
                                              float* __restrict__ msg) {
  extern __shared__ float smem[];
  float* fL = smem;                  // [EE][K1P]
  float* wL = smem + EE * K1P;       // [HH][K1P]
  const int b = blockIdx.x, tid = threadIdx.x;

  stage_rows4(Wi, wL, HH, FD, K1P, tid);
  zero_pad(wL, HH, FD, K1P, K1P - FD, tid);
  const float* fB = fini + (long)b * EE * FD;
  stage_rows4(fB, fL, EE, FD, K1P, tid);
  zero_pad(fL, EE, FD, K1P, K1P - FD, tid);
  __syncthreads();

  const int lane = tid & 31, wave = tid >> 5;
  const int l15 = lane & 15, hi2 = (lane >> 4) << 1;  // 0 or 2
  const int n = wave * 16 + l15;                      // output column (h)
  const float* brow = wL + n * K1P;
  float* inpB = inp + (long)b * EE * HH;
  float* msgB = msg + (long)b * EE * HH;

  for (int mt = 0; mt < 16; ++mt) {
    const float* arow = fL + (mt * 16 + l15) * K1P;
    v8f acc = {};
    for (int k0 = 0; k0 < K1P; k0 += 4) {
      v2f a, bb;
      a.x = arow[k0 + hi2];  a.y = arow[k0 + hi2 + 1];
      bb.x = brow[k0 + hi2]; bb.y = brow[k0 + hi2 + 1];
      acc = wmma4(a, bb, acc);
    }
#pragma unroll
    for (int v = 0; v < 8; ++v) {
      int e = mt * 16 + v + (hi2 << 2);  // +8 for upper half-wave
      long idx = (long)e * HH + n;
      float val = acc[v];
      inpB[idx] = val;
      msgB[idx] = fmaxf(val, 0.f);
    }
  }
}

// ---------------------------------------------------------------------------
// Kernel 2: one message-passing iteration (in-place msg update).
//   agg[e] = sum_k msg[map[e,k]]   (gathered into LDS)
//   msg    = relu(inp + agg @ W_h^T)
// LDS: agg 256x132 + W_h 128x132 + mapping 1536 ints
// ---------------------------------------------------------------------------
__global__ void __launch_bounds__(256) k_mp(const int* __restrict__ mapping,
                                            const float* __restrict__ Wh,
                                            const float* __restrict__ inp,
                                            float* __restrict__ msg) {
  extern __shared__ float smem[];
  float* aggL = smem;                        // [EE][S2]
  float* wL   = aggL + EE * S2;              // [HH][S2]
  int*   mapL = (int*)(wL + HH * S2);        // [EE*KK]
  const int b = blockIdx.x, tid = threadIdx.x;

  stage_rows4(Wh, wL, HH, HH, S2, tid);  // 128 cols: never straddles
  for (int i = tid; i < EE * KK; i += 256)
    mapL[i] = mapping[(long)b * EE * KK + i];
  __syncthreads();

  float* msgB = msg + (long)b * EE * HH;
  const float* inpB = inp + (long)b * EE * HH;
  for (int t = tid; t < EE * (HH / 4); t += 256) {
    int e = t >> 5, c4 = (t & 31) << 2;
    float sx = 0.f, sy = 0.f, sz = 0.f, sw = 0.f;
#pragma unroll
    for (int k = 0; k < KK; ++k) {
      float4 m4 = *(const float4*)(msgB + mapL[e * KK + k] * HH + c4);
      sx += m4.x; sy += m4.y; sz += m4.z; sw += m4.w;
    }
    float* d = aggL + e * S2 + c4;
    d[0] = sx; d[1] = sy; d[2] = sz; d[3] = sw;
  }
  __syncthreads();  // all gathers done before msg is overwritten

  const int lane = tid & 31, wave = tid >> 5;
  const int l15 = lane & 15, hi2 = (lane >> 4) << 1;
  const int n = wave * 16 + l15;
  const float* brow = wL + n * S2;

  for (int mt = 0; mt < 16; ++mt) {
    const float* arow = aggL + (mt * 16 + l15) * S2;
    v8f acc = {};
    for (int k0 = 0; k0 < HH; k0 += 4) {
      v2f a, bb;
      a.x = arow[k0 + hi2];  a.y = arow[k0 + hi2 + 1];
      bb.x = brow[k0 + hi2]; bb.y = brow[k0 + hi2 + 1];
      acc = wmma4(a, bb, acc);
    }
#pragma unroll
    for (int v = 0; v < 8; ++v) {
      int e = mt * 16 + v + (hi2 << 2);
      long idx = (long)e * HH + n;
      msgB[idx] = fmaxf(acc[v] + inpB[idx], 0.f);
    }
  }
}

// ---------------------------------------------------------------------------
// Kernel 3: readout.
//   m2a[a] = sum_k msg[a2b[a,k]] ; cat = [atom_feat | m2a] (in LDS, 268 wide)
//   atoms_h = relu(cat @ W_o^T + b_o); scatter valid rows to comp; write mask.
// LDS: cat 128x268 + W_o 128x268 + a2b 768 ints
// ---------------------------------------------------------------------------
__global__ void __launch_bounds__(256) k_out(const float* __restrict__ atomf,
                                             const int* __restrict__ a2b,
                                             const float* __restrict__ Wo,
                                             const float* __restrict__ bo,
                                             const float* __restrict__ msg,
                                             const int* __restrict__ lens,
                                             const int* __restrict__ prefix,
                                             float* __restrict__ out) {
  extern __shared__ float smem[];
  float* catL = smem;                        // [AA][CATP]
  float* wL   = catL + AA * CATP;            // [HH][CATP]
  int*   abL  = (int*)(wL + HH * CATP);      // [AA*KK]
  const int b = blockIdx.x, tid = threadIdx.x;

  stage_rows4(Wo, wL, HH, CATD, CATP, tid);
  zero_pad(wL, HH, CATD, CATP, CATP - CATD, tid);
  for (int i = tid; i < AA * KK; i += 256)
    abL[i] = a2b[(long)b * AA * KK + i];
  const float* afB = atomf + (long)b * AA * AF;
  stage_rows4(afB, catL, AA, AF, CATP, tid);   // cols 0..132 of cat
  zero_pad(catL, AA, CATD, CATP, CATP - CATD, tid);
  __syncthreads();  // abL ready

  const float* msgB = msg + (long)b * EE * HH;
  for (int t = tid; t < AA * (HH / 4); t += 256) {
    int a = t >> 5, c4 = (t & 31) << 2;
    float sx = 0.f, sy = 0.f, sz = 0.f, sw = 0.f;
#pragma unroll
    for (int k = 0; k < KK; ++k) {
      float4 m4 = *(const float4*)(msgB + abL[a * KK + k] * HH + c4);
      sx += m4.x; sy += m4.y; sz += m4.z; sw += m4.w;
    }
    float* d = catL + a * CATP + AF + c4;  // m2a block (base unaligned -> scalar)
    d[0] = sx; d[1] = sy; d[2] = sz; d[3] = sw;
  }
  __syncthreads();

  const int lane = tid & 31, wave = tid >> 5;
  const int l15 = lane & 15, hi2 = (lane >> 4) << 1;
  const int n = wave * 16 + l15;
  const float bias = bo[n];
  const float* brow = wL + n * CATP;
  const int lenb = lens[b];
  const long base = prefix[b];
  float* cmask = out + (long)prefix[BB] * HH;  // mask region after comp

  for (int mt = 0; mt < 8; ++mt) {
    const float* arow = catL + (mt * 16 + l15) * CATP;
    v8f acc = {};
    for (int k0 = 0; k0 < CATP; k0 += 4) {
      v2f a, bb;
      a.x = arow[k0 + hi2];  a.y = arow[k0 + hi2 + 1];
      bb.x = brow[k0 + hi2]; bb.y = brow[k0 + hi2 + 1];
      acc = wmma4(a, bb, acc);
    }
#pragma unroll
    for (int v = 0; v < 8; ++v) {
      int a = mt * 16 + v + (hi2 << 2);
      float val = fmaxf(acc[v] + bias, 0.f);
      if (a < lenb) out[(base + a) * (long)HH + n] = val;
    }
  }
  if (tid < AA) cmask[(long)b * AA + tid] = (tid < lenb) ? 1.f : 0.f;
}

// ---------------------------------------------------------------------------
// Host launcher
// ---------------------------------------------------------------------------
extern "C" void kernel_launch(void* const* d_in, const int* in_sizes, int n_in,
                              void* d_out, int out_size, void* d_ws, size_t ws_size,
                              hipStream_t stream) {
  const float* atomf   = (const float*)d_in[0];
  const float* fini    = (const float*)d_in[1];
  const int*   a2b     = (const int*)d_in[2];
  const int*   mapping = (const int*)d_in[3];
  // d_in[4] = global_f: unused by the reference output
  const int*   lens    = (const int*)d_in[5];
  const float* Wi      = (const float*)d_in[6];
  const float* Wh      = (const float*)d_in[7];
  const float* Wo      = (const float*)d_in[8];
  const float* bo      = (const float*)d_in[9];

  float* ws  = (float*)d_ws;
  float* inp = ws;                              // [BB*EE*HH]
  float* msg = ws + (long)BB * EE * HH;         // [BB*EE*HH]
  int* prefix = (int*)(ws + 2L * BB * EE * HH); // [BB+1]

  const size_t s1 = (size_t)(EE + HH) * K1P * sizeof(float);
  const size_t s2 = (size_t)(EE + HH) * S2 * sizeof(float) + EE * KK * sizeof(int);
  const size_t s3 = (size_t)(AA + HH) * CATP * sizeof(float) + AA * KK * sizeof(int);
  hipFuncSetAttribute((const void*)k_init, hipFuncAttributeMaxDynamicSharedMemorySize, (int)s1);
  hipFuncSetAttribute((const void*)k_mp,   hipFuncAttributeMaxDynamicSharedMemorySize, (int)s2);
  hipFuncSetAttribute((const void*)k_out,  hipFuncAttributeMaxDynamicSharedMemorySize, (int)s3);

  k_prefix<<<1, 32, 0, stream>>>(lens, prefix);
  k_init<<<BB, 256, s1, stream>>>(fini, Wi, inp, msg);
  k_mp<<<BB, 256, s2, stream>>>(mapping, Wh, inp, msg);   // depth iter 1
  k_mp<<<BB, 256, s2, stream>>>(mapping, Wh, inp, msg);   // depth iter 2
  k_out<<<BB, 256, s3, stream>>>(atomf, a2b, Wo, bo, msg, lens, prefix,
                                 (float*)d_out);
}